// LightGCN_4269197492541
// MI455X (gfx1250) — compile-verified
//
#include <hip/hip_runtime.h>
#include <cstdint>

#define N_USERS_C 100000
#define N_ITEMS_C 50000
#define N_NODES_C 150000
#define EMB_DIM_C 64

// ---------------------------------------------------------------------------
// CDNA5 async global -> LDS copy (gfx1250). VDST operand = 32-bit LDS byte
// address (low 32 bits of the flat pointer), VADDR = 64-bit global address.
// Edge stream has zero reuse -> non-temporal hint so it does not evict the
// L2-resident embedding tables (~115 MB working set vs 192 MB L2).
// Tracked with ASYNCcnt; wave-local completion via s_wait_asynccnt.
// ---------------------------------------------------------------------------
__device__ __forceinline__ void async_ld_b32_nt(uint32_t lds_addr, const void* gptr) {
    asm volatile("global_load_async_to_lds_b32 %0, %1, off th:TH_LOAD_NT"
                 :: "v"(lds_addr), "v"(gptr)
                 : "memory");
}

__device__ __forceinline__ void wait_asynccnt0() {
#if __has_builtin(__builtin_amdgcn_s_wait_asynccnt)
    __builtin_amdgcn_s_wait_asynccnt(0);
#else
    asm volatile("s_wait_asynccnt 0" ::: "memory");
#endif
}

// Force the hardware no-return global_atomic_add_f32 (avoid any CAS fallback).
__device__ __forceinline__ void atomic_add_f32(float* p, float v) {
#if defined(__HIP_DEVICE_COMPILE__)
    (void)unsafeAtomicAdd(p, v);
#else
    (void)atomicAdd(p, v);
#endif
}

// ---------------------------------------------------------------------------
// SpMM: dst[row] += val * src[col]  (scatter-add over 4.8M edges)
// 256 threads = 8 waves per block; each wave owns 32 edges:
//   * stage (row,col,val) for its 32 edges into its LDS slice asynchronously
//   * then 16 rounds x 2 edges: 16 lanes per edge, float4 per lane
// Source row selected via (col < N_USERS) between two base pointers so the
// same kernel handles the split user/item layer-0 inputs and the
// contiguous workspace buffers of later layers.
// ---------------------------------------------------------------------------
__global__ __launch_bounds__(256) void spmm_scatter_kernel(
    const int* __restrict__ rows, const int* __restrict__ cols,
    const float* __restrict__ vals,
    const float* __restrict__ srcU, const float* __restrict__ srcI,
    float* __restrict__ dst, int nEdges)
{
    __shared__ int   s_rows[256];
    __shared__ int   s_cols[256];
    __shared__ float s_vals[256];

    const int tid       = threadIdx.x;
    const int blockBase = blockIdx.x * 256;

    // Stage this thread's edge triple into LDS (clamped tail -> harmless dup).
    int e  = blockBase + tid;
    int ee = (e < nEdges) ? e : (nEdges - 1);
    async_ld_b32_nt((uint32_t)(uintptr_t)&s_rows[tid], rows + ee);
    async_ld_b32_nt((uint32_t)(uintptr_t)&s_cols[tid], cols + ee);
    async_ld_b32_nt((uint32_t)(uintptr_t)&s_vals[tid], vals + ee);
    wait_asynccnt0();   // wave-local: consumers below are the same wave's lanes

    const int lane     = tid & 31;
    const int waveSlot = tid & ~31;   // this wave's 32-edge LDS slice
    const int sub      = lane & 15;   // 16 lanes cover 64 floats as float4
    const int grp      = lane >> 4;   // which of the 2 edges this round

    #pragma unroll 4
    for (int r = 0; r < 16; ++r) {
        const int li = waveSlot + r * 2 + grp;
        const int ge = blockBase + li;
        if (ge < nEdges) {
            const int   row = s_rows[li];
            const int   col = s_cols[li];
            const float v   = s_vals[li];
            const float* src = (col < N_USERS_C)
                ? (srcU + (size_t)col * EMB_DIM_C)
                : (srcI + (size_t)(col - N_USERS_C) * EMB_DIM_C);
            const float4 x = *(const float4*)(src + sub * 4);   // coalesced b128
            float* d = dst + (size_t)row * EMB_DIM_C + sub * 4;
            // no-return f32 atomics, resolved in L2 (table is L2-resident)
            atomic_add_f32(d + 0, v * x.x);
            atomic_add_f32(d + 1, v * x.y);
            atomic_add_f32(d + 2, v * x.z);
            atomic_add_f32(d + 3, v * x.w);
        }
    }
}

// ---------------------------------------------------------------------------
// Elementwise helpers (float4-vectorized, one element per thread)
// ---------------------------------------------------------------------------

// acc = concat(user_emb, item_emb); buf = 0
__global__ __launch_bounds__(256) void init_kernel(
    const float* __restrict__ u, const float* __restrict__ it,
    float* __restrict__ acc, float* __restrict__ buf, int totalQ)
{
    int i = blockIdx.x * blockDim.x + threadIdx.x;
    if (i >= totalQ) return;
    const int userQ = N_USERS_C * EMB_DIM_C / 4;
    float4 v = (i < userQ) ? ((const float4*)u)[i] : ((const float4*)it)[i - userQ];
    ((float4*)acc)[i] = v;
    ((float4*)buf)[i] = make_float4(0.f, 0.f, 0.f, 0.f);
}

// acc += lay; z = 0   (z is the buffer receiving the NEXT layer's scatter)
__global__ __launch_bounds__(256) void addzero_kernel(
    float* __restrict__ acc, const float* __restrict__ lay,
    float* __restrict__ z, int totalQ)
{
    int i = blockIdx.x * blockDim.x + threadIdx.x;
    if (i >= totalQ) return;
    float4 a = ((const float4*)acc)[i];
    float4 l = ((const float4*)lay)[i];
    a.x += l.x; a.y += l.y; a.z += l.z; a.w += l.w;
    ((float4*)acc)[i] = a;
    ((float4*)z)[i]   = make_float4(0.f, 0.f, 0.f, 0.f);
}

// acc = (acc + lay) * 0.25
__global__ __launch_bounds__(256) void final_kernel(
    float* __restrict__ acc, const float* __restrict__ lay, int totalQ)
{
    int i = blockIdx.x * blockDim.x + threadIdx.x;
    if (i >= totalQ) return;
    float4 a = ((const float4*)acc)[i];
    float4 l = ((const float4*)lay)[i];
    a.x = (a.x + l.x) * 0.25f;
    a.y = (a.y + l.y) * 0.25f;
    a.z = (a.z + l.z) * 0.25f;
    a.w = (a.w + l.w) * 0.25f;
    ((float4*)acc)[i] = a;
}

// ---------------------------------------------------------------------------
// Launch: acc(d_out) = e0; for k=1..3: e_k = A e_{k-1} (scatter into zeroed
// ping-pong buffer), acc += e_k; final acc *= 1/4.
// Workspace: 2 x N*D floats = 76.8 MB.
// ---------------------------------------------------------------------------
extern "C" void kernel_launch(void* const* d_in, const int* in_sizes, int n_in,
                              void* d_out, int out_size, void* d_ws, size_t ws_size,
                              hipStream_t stream) {
    const int*   rows = (const int*)d_in[0];
    const int*   cols = (const int*)d_in[1];
    const float* vals = (const float*)d_in[2];
    const float* uemb = (const float*)d_in[3];
    const float* iemb = (const float*)d_in[4];

    float* acc  = (float*)d_out;
    float* bufA = (float*)d_ws;
    float* bufB = bufA + (size_t)N_NODES_C * EMB_DIM_C;

    const int nEdges = in_sizes[0];
    const int totalQ = N_NODES_C * EMB_DIM_C / 4;     // float4 count
    const int ewBlocks   = (totalQ + 255) / 256;
    const int spmmBlocks = (nEdges + 255) / 256;

    const size_t uOff = (size_t)N_USERS_C * EMB_DIM_C;

    // acc = e0 ; bufA = 0
    init_kernel<<<ewBlocks, 256, 0, stream>>>(uemb, iemb, acc, bufA, totalQ);

    // layer 1: bufA = A * e0 (read split user/item inputs)
    spmm_scatter_kernel<<<spmmBlocks, 256, 0, stream>>>(
        rows, cols, vals, uemb, iemb, bufA, nEdges);
    // acc += e1 ; bufB = 0
    addzero_kernel<<<ewBlocks, 256, 0, stream>>>(acc, bufA, bufB, totalQ);

    // layer 2: bufB = A * e1
    spmm_scatter_kernel<<<spmmBlocks, 256, 0, stream>>>(
        rows, cols, vals, bufA, bufA + uOff, bufB, nEdges);
    // acc += e2 ; bufA = 0
    addzero_kernel<<<ewBlocks, 256, 0, stream>>>(acc, bufB, bufA, totalQ);

    // layer 3: bufA = A * e2
    spmm_scatter_kernel<<<spmmBlocks, 256, 0, stream>>>(
        rows, cols, vals, bufB, bufB + uOff, bufA, nEdges);

    // acc = (acc + e3) / 4
    final_kernel<<<ewBlocks, 256, 0, stream>>>(acc, bufA, totalQ);
}